// SCRF_43069932045121
// MI455X (gfx1250) — compile-verified
//
#include <hip/hip_runtime.h>

typedef __attribute__((ext_vector_type(2))) float v2f;
typedef __attribute__((ext_vector_type(8))) float v8f;

#define NSEQ   512
#define DDIM   256
#define NTAG   12
#define SPANW  8
#define NEGV   (-1e30f)

// ---------------------------------------------------------------------------
// Stage a 16-row weight slab (16 x K fp32) into padded LDS using the gfx1250
// async copy path: GLOBAL_LOAD_ASYNC_TO_LDS_B128 (tracked by ASYNCcnt).
// Thread t -> row n = t/16, chunk lane c0 = t%16; each chunk = 16 bytes.
// Caller must __syncthreads() afterwards to publish to all waves.
// ---------------------------------------------------------------------------
__device__ __forceinline__ void stage_tile_async(const float* __restrict__ wrows,
                                                 int ldw, int K,
                                                 float* lds_row0, int lstride,
                                                 int tid) {
  int n  = tid >> 4;          // 0..15
  int c0 = tid & 15;
  const float* g = wrows + (size_t)n * ldw;
  unsigned lbase = (unsigned)(uintptr_t)(lds_row0 + (size_t)n * lstride);
  int nchunks = K >> 2;       // b128 chunks per row
  for (int c = c0; c < nchunks; c += 16) {
    unsigned long long ga = (unsigned long long)(uintptr_t)(g + 4 * c);
    unsigned la = lbase + 16u * c;
    asm volatile("global_load_async_to_lds_b128 %0, %1, off"
                 :: "v"(la), "v"(ga)
                 : "memory");
  }
  asm volatile("s_wait_asynccnt 0x0" ::: "memory");
}

// ---------------------------------------------------------------------------
// 1) ht0 = feats(512x512) @ dense_w^T(512x256) + dense_b  -> BAND[0]
//    Grid: 32 WGs = 16 N-tiles x 2 M-groups; 8 waves/WG; wave = 32-row M-pair.
// ---------------------------------------------------------------------------
__global__ void scrf_dense(const float* __restrict__ feats,
                           const float* __restrict__ w,
                           const float* __restrict__ bias,
                           float* __restrict__ out) {
  __shared__ float Wl[16][516];            // K=512, pad 4 -> bank stride 4
  int tid  = threadIdx.x;
  int n0   = (blockIdx.x & 15) << 4;
  int mgrp = blockIdx.x >> 4;              // 0..1
  stage_tile_async(w + (size_t)n0 * 512, 512, 512, &Wl[0][0], 516, tid);
  __syncthreads();

  int wave = tid >> 5, lane = tid & 31;
  int h = lane >> 4, l = lane & 15;
  int p  = mgrp * 8 + wave;                // M pair 0..15 (32 rows each)
  int rb = p << 5;
  const float* a0 = feats + (size_t)(rb + l) * 512;
  const float* a1 = a0 + 16 * 512;
  v8f acc0 = {}, acc1 = {};
  #pragma unroll 4
  for (int k0 = 0; k0 < 512; k0 += 4) {
    int kk = k0 + 2 * h;
    v2f av0; av0.x = a0[kk];    av0.y = a0[kk + 1];
    v2f av1; av1.x = a1[kk];    av1.y = a1[kk + 1];
    v2f bv;  bv.x  = Wl[l][kk]; bv.y  = Wl[l][kk + 1];
    acc0 = __builtin_amdgcn_wmma_f32_16x16x4_f32(false, av0, false, bv,
                                                 (short)0, acc0, false, false);
    acc1 = __builtin_amdgcn_wmma_f32_16x16x4_f32(false, av1, false, bv,
                                                 (short)0, acc1, false, false);
  }
  float bvv = bias[n0 + l];
  #pragma unroll
  for (int v = 0; v < 8; ++v) {
    int row = rb + v + 8 * h;
    out[(size_t)row * DDIM + n0 + l]        = acc0[v] + bvv;
    out[(size_t)(row + 16) * DDIM + n0 + l] = acc1[v] + bvv;
  }
}

// ---------------------------------------------------------------------------
// 2a) P(len x 2048) = ht(len x 256) @ [wl_w; wr_w; gl_w; gr_w]^T
//     Grid: 128 WGs (one per N-tile). LDS-staged weight slab shared by 8
//     waves; each wave walks two 32-row M-pairs with dual accumulators.
// ---------------------------------------------------------------------------
__global__ void scrf_cat_gemm(const float* __restrict__ ht, int len,
                              const float* __restrict__ wl,
                              const float* __restrict__ wr,
                              const float* __restrict__ gl,
                              const float* __restrict__ gr,
                              float* __restrict__ P) {
  __shared__ float Wl[16][260];            // K=256, pad 4
  int tid = threadIdx.x;
  int n0  = blockIdx.x << 4;
  const float* W; int nl;
  if (n0 < 256)       { W = wl; nl = n0;        }
  else if (n0 < 512)  { W = wr; nl = n0 - 256;  }
  else if (n0 < 1280) { W = gl; nl = n0 - 512;  }
  else                { W = gr; nl = n0 - 1280; }
  stage_tile_async(W + (size_t)nl * DDIM, DDIM, DDIM, &Wl[0][0], 260, tid);
  __syncthreads();

  int wave = tid >> 5, lane = tid & 31;
  int h = lane >> 4, l = lane & 15;
  for (int p = wave; p < 16; p += 8) {     // two M-pairs per wave
    int rb = p << 5;
    int r0 = rb + l;      if (r0 >= len) r0 = len - 1;   // branchless clamp
    int r1 = rb + 16 + l; if (r1 >= len) r1 = len - 1;
    const float* a0 = ht + (size_t)r0 * DDIM;
    const float* a1 = ht + (size_t)r1 * DDIM;
    v8f acc0 = {}, acc1 = {};
    #pragma unroll 4
    for (int k0 = 0; k0 < DDIM; k0 += 4) {
      int kk = k0 + 2 * h;
      v2f av0; av0.x = a0[kk];    av0.y = a0[kk + 1];
      v2f av1; av1.x = a1[kk];    av1.y = a1[kk + 1];
      v2f bv;  bv.x  = Wl[l][kk]; bv.y  = Wl[l][kk + 1];
      acc0 = __builtin_amdgcn_wmma_f32_16x16x4_f32(false, av0, false, bv,
                                                   (short)0, acc0, false, false);
      acc1 = __builtin_amdgcn_wmma_f32_16x16x4_f32(false, av1, false, bv,
                                                   (short)0, acc1, false, false);
    }
    #pragma unroll
    for (int v = 0; v < 8; ++v) {
      int row = rb + v + 8 * h;
      if (row < len)      P[(size_t)row * 2048 + n0 + l]        = acc0[v];
      if (row + 16 < len) P[(size_t)(row + 16) * 2048 + n0 + l] = acc1[v];
    }
  }
}

// ---------------------------------------------------------------------------
// 2b) gated combine: ht_out[i] = softmax(g)·{4σ(pl+pr)-2, hl, hr}
// ---------------------------------------------------------------------------
__global__ void scrf_combine(const float* __restrict__ P,
                             const float* __restrict__ ht_in,
                             const float* __restrict__ wlb,
                             const float* __restrict__ wrb,
                             const float* __restrict__ glb,
                             const float* __restrict__ grb,
                             float* __restrict__ ht_out, int len_out) {
  int i = blockIdx.x;
  int d = threadIdx.x;
  if (i >= len_out) return;
  const float* Pi = P + (size_t)i * 2048;
  const float* Pj = P + (size_t)(i + 1) * 2048;
  float sh   = Pi[d] + wlb[d] + Pj[256 + d] + wrb[d];
  float hhat = 4.0f / (1.0f + __expf(-sh)) - 2.0f;
  float g0 = Pi[512  + d] + Pj[1280 + d] + glb[d]       + grb[d];
  float g1 = Pi[768  + d] + Pj[1536 + d] + glb[256 + d] + grb[256 + d];
  float g2 = Pi[1024 + d] + Pj[1792 + d] + glb[512 + d] + grb[512 + d];
  float mg = fmaxf(g0, fmaxf(g1, g2));
  float e0 = __expf(g0 - mg), e1 = __expf(g1 - mg), e2 = __expf(g2 - mg);
  float inv = 1.0f / (e0 + e1 + e2);
  float hl = ht_in[(size_t)i * DDIM + d];
  float hr = ht_in[(size_t)(i + 1) * DDIM + d];
  ht_out[(size_t)i * DDIM + d] = (e0 * hhat + e1 * hl + e2 * hr) * inv;
}

// ---------------------------------------------------------------------------
// 3) TP(8*512 x 16) = BAND(4096 x 256) @ tag_w^T (12 valid cols, 4 zero-pad)
// ---------------------------------------------------------------------------
__global__ void scrf_tagproj(const float* __restrict__ band,
                             const float* __restrict__ tw,
                             const float* __restrict__ tb,
                             float* __restrict__ tp) {
  int wv   = (blockIdx.x * blockDim.x + threadIdx.x) >> 5;
  int lane = threadIdx.x & 31;
  int h = lane >> 4, l = lane & 15;
  int m0 = wv << 4;                       // 256 M tiles, 1 N tile
  const float* arow = band + (size_t)(m0 + l) * DDIM;
  const float* brow = tw + (size_t)(l < NTAG ? l : 0) * DDIM;
  float msk = (l < NTAG) ? 1.0f : 0.0f;   // zero-pad tags 12..15
  v8f acc = {};
  #pragma unroll 4
  for (int k0 = 0; k0 < DDIM; k0 += 4) {
    int kk = k0 + 2 * h;
    v2f a; a.x = arow[kk];       a.y = arow[kk + 1];
    v2f b; b.x = brow[kk] * msk; b.y = brow[kk + 1] * msk;
    acc = __builtin_amdgcn_wmma_f32_16x16x4_f32(false, a, false, b,
                                                (short)0, acc, false, false);
  }
  if (l < NTAG) {
    float bv = tb[l];
    #pragma unroll
    for (int v = 0; v < 8; ++v) {
      int row = m0 + v + 8 * h;
      tp[(size_t)row * 16 + l] = acc[v] + bv;
    }
  }
}

// ---------------------------------------------------------------------------
// 4) sequential band log-partition scan (single workgroup, 512 steps)
// ---------------------------------------------------------------------------
__global__ void scrf_scan(const float* __restrict__ tp,
                          const float* __restrict__ trans,
                          float* __restrict__ partition) {
  __shared__ float buf[SPANW][NTAG];
  __shared__ float inner[NTAG];
  __shared__ float tr[NTAG * NTAG];
  int tid = threadIdx.x;
  if (tid < NTAG) {
    for (int k = 0; k < SPANW; ++k) buf[k][tid] = NEGV;
    buf[SPANW - 1][tid] = (tid == 10) ? 0.0f : -1000.0f;   // alpha0, START=10
  }
  for (int q = tid; q < NTAG * NTAG; q += blockDim.x) tr[q] = trans[q];
  __syncthreads();
  for (int j = 0; j < NSEQ; ++j) {
    if (tid < NTAG) {                    // inner[t1] = lse_k(buf[k]+tp[s,i,t1])
      float vals[SPANW]; float mx = -3.4e38f;
      #pragma unroll
      for (int k = 0; k < SPANW; ++k) {
        int i = j - 7 + k; if (i < 0) i = 0;
        int s = j - i;
        float v = buf[k][tid] + tp[((size_t)s * NSEQ + i) * 16 + tid];
        vals[k] = v; mx = fmaxf(mx, v);
      }
      float sum = 0.0f;
      #pragma unroll
      for (int k = 0; k < SPANW; ++k) sum += __expf(vals[k] - mx);
      inner[tid] = mx + __logf(sum);
    }
    __syncthreads();
    float nv = 0.0f;
    if (tid < NTAG) {                    // alpha[t2] = lse_t1(inner+trans)
      float vals[NTAG]; float mx = -3.4e38f;
      #pragma unroll
      for (int t1 = 0; t1 < NTAG; ++t1) {
        float v = inner[t1] + tr[t1 * NTAG + tid];
        vals[t1] = v; mx = fmaxf(mx, v);
      }
      float sum = 0.0f;
      #pragma unroll
      for (int t1 = 0; t1 < NTAG; ++t1) sum += __expf(vals[t1] - mx);
      nv = mx + __logf(sum);
    }
    __syncthreads();
    if (tid < NTAG) {                    // shift window
      for (int k = 0; k < SPANW - 1; ++k) buf[k][tid] = buf[k + 1][tid];
      buf[SPANW - 1][tid] = nv;
    }
    __syncthreads();
  }
  if (tid == 0) {
    float mx = -3.4e38f;
    for (int t = 0; t < NTAG; ++t) mx = fmaxf(mx, buf[SPANW - 1][t]);
    float sum = 0.0f;
    for (int t = 0; t < NTAG; ++t) sum += __expf(buf[SPANW - 1][t] - mx);
    partition[0] = mx + __logf(sum);
  }
}

// ---------------------------------------------------------------------------
// 5) gold gather (64 spans) + final scalar
// ---------------------------------------------------------------------------
__global__ void scrf_gold(const int* __restrict__ tags,
                          const float* __restrict__ tp,
                          const float* __restrict__ tb,
                          const float* __restrict__ trans,
                          const float* __restrict__ partition,
                          float* __restrict__ out) {
  __shared__ float red[64];
  int m = threadIdx.x;
  int a  = tags[m * 4 + 0];
  int b  = tags[m * 4 + 1];
  int c  = tags[m * 4 + 2];
  int d2 = tags[m * 4 + 3];
  int s = b - a;
  float tpv = (s >= 0 && s < SPANW) ? tp[((size_t)s * NSEQ + a) * 16 + c]
                                    : tb[c];   // H==0 off the band
  red[m] = tpv + trans[c * NTAG + d2];
  __syncthreads();
  for (int off = 32; off > 0; off >>= 1) {
    if (m < off) red[m] += red[m + off];
    __syncthreads();
  }
  if (m == 0) out[0] = partition[0] - red[0];
}

// ---------------------------------------------------------------------------
extern "C" void kernel_launch(void* const* d_in, const int* in_sizes, int n_in,
                              void* d_out, int out_size, void* d_ws, size_t ws_size,
                              hipStream_t stream) {
  const float* feats   = (const float*)d_in[0];
  const int*   tags    = (const int*)  d_in[1];
  const float* dense_w = (const float*)d_in[2];
  const float* dense_b = (const float*)d_in[3];
  const float* wl_w    = (const float*)d_in[4];
  const float* wl_b    = (const float*)d_in[5];
  const float* wr_w    = (const float*)d_in[6];
  const float* wr_b    = (const float*)d_in[7];
  const float* gl_w    = (const float*)d_in[8];
  const float* gl_b    = (const float*)d_in[9];
  const float* gr_w    = (const float*)d_in[10];
  const float* gr_b    = (const float*)d_in[11];
  const float* tag_w   = (const float*)d_in[12];
  const float* tag_b   = (const float*)d_in[13];
  const float* trans   = (const float*)d_in[14];

  // workspace layout (floats): BAND[8][512][256] | P[512][2048] | TP[8*512][16] | PART
  float* BAND = (float*)d_ws;
  float* P    = BAND + (size_t)SPANW * NSEQ * DDIM;
  float* TP   = P    + (size_t)NSEQ * 2048;
  float* PART = TP   + (size_t)SPANW * NSEQ * 16;

  // 1) ht0 -> BAND[0]
  scrf_dense<<<32, 256, 0, stream>>>(feats, dense_w, dense_b, BAND);

  // 2) recurrence s = 1..7 (sequential dependency chain)
  for (int s = 1; s < SPANW; ++s) {
    int len_in  = NSEQ - (s - 1);
    int len_out = NSEQ - s;
    const float* htin  = BAND + (size_t)(s - 1) * NSEQ * DDIM;
    float*       htout = BAND + (size_t)s       * NSEQ * DDIM;
    scrf_cat_gemm<<<128, 256, 0, stream>>>(htin, len_in, wl_w, wr_w, gl_w, gr_w, P);
    scrf_combine<<<len_out, 256, 0, stream>>>(P, htin, wl_b, wr_b, gl_b, gr_b,
                                              htout, len_out);
  }

  // 3) tag projection of all 8 diagonals (256 waves)
  scrf_tagproj<<<32, 256, 0, stream>>>(BAND, tag_w, tag_b, TP);

  // 4) log-partition scan (single workgroup)
  scrf_scan<<<1, 64, 0, stream>>>(TP, trans, PART);

  // 5) gold score + final output
  scrf_gold<<<1, 64, 0, stream>>>(tags, TP, tag_b, trans, PART, (float*)d_out);
}